// MPAtomCenteredDescriptor_76622216560726
// MI455X (gfx1250) — compile-verified
//
#include <hip/hip_runtime.h>
#include <math.h>

// ---------------- constants describing the problem ----------------
#define RDIM 16      // NUM_RADIAL
#define FDIM 16      // F
#define CH   16      // 16 irrep channels (l=0..3)
#define NPATH 17
#define CG_TOTAL 1077
#define CUTOFF_R 5.0f

constexpr int PATH_L1[NPATH] = {0,0,0,1,1,1,1,1,2,2,2,2,2,3,3,3,3};
constexpr int PATH_L2[NPATH] = {0,1,2,0,1,1,1,2,0,1,1,2,2,0,1,2,2};
constexpr int PATH_L3[NPATH] = {0,1,2,1,0,2,1,3,2,1,3,0,2,3,2,1,3};
constexpr int PATH_OFF[NPATH] = {0,1,10,35,44,53,98,143,248,273,318,423,448,573,622,727,832};

typedef __attribute__((ext_vector_type(2))) float v2f;
typedef __attribute__((ext_vector_type(8))) float v8f;

// ---------------- utility kernels ----------------
__global__ void fill_zero_kernel(float* p, int n) {
    int t = blockIdx.x * blockDim.x + threadIdx.x;
    if (t < n) p[t] = 0.0f;
}

__global__ void copy_kernel(float* dst, const float* src, int n) {
    int t = blockIdx.x * blockDim.x + threadIdx.x;
    if (t < n) dst[t] = src[t];
}

// ---------------- Clebsch-Gordan init (device, double precision) ----------------
__device__ double dfact(int n) { double r = 1.0; for (int i = 2; i <= n; i++) r *= (double)i; return r; }

__device__ double cg_coef(int j1, int m1, int j2, int m2, int j3, int m3) {
    if (m1 + m2 != m3) return 0.0;
    if (j3 < abs(j1 - j2) || j3 > j1 + j2) return 0.0;
    double pre = sqrt((2.0 * j3 + 1.0) * dfact(j3 + j1 - j2) * dfact(j3 - j1 + j2) *
                      dfact(j1 + j2 - j3) / dfact(j1 + j2 + j3 + 1));
    pre *= sqrt(dfact(j3 + m3) * dfact(j3 - m3) * dfact(j1 - m1) * dfact(j1 + m1) *
                dfact(j2 - m2) * dfact(j2 + m2));
    double s = 0.0;
    for (int k = 0; k <= j1 + j2 - j3; k++) {
        int d0 = k, d1 = j1 + j2 - j3 - k, d2 = j1 - m1 - k;
        int d3 = j2 + m2 - k, d4 = j3 - j2 + m1 + k, d5 = j3 - j1 - m2 + k;
        if (d0 < 0 || d1 < 0 || d2 < 0 || d3 < 0 || d4 < 0 || d5 < 0) continue;
        double denom = dfact(d0) * dfact(d1) * dfact(d2) * dfact(d3) * dfact(d4) * dfact(d5);
        s += ((k & 1) ? -1.0 : 1.0) / denom;
    }
    return pre * s;
}

__device__ void build_u(int l, double Ur[7][7], double Ui[7][7]) {
    for (int a = 0; a < 7; a++) for (int b = 0; b < 7; b++) { Ur[a][b] = 0.0; Ui[a][b] = 0.0; }
    Ur[l][l] = 1.0;
    double is2 = 1.0 / sqrt(2.0);
    for (int m = 1; m <= l; m++) {
        double sgn = (m & 1) ? -1.0 : 1.0;
        Ur[l + m][l + m] = sgn * is2;
        Ur[l + m][l - m] = is2;
        Ui[l - m][l - m] = is2;
        Ui[l - m][l + m] = -sgn * is2;
    }
}

__global__ void init_cg_kernel(float* cg) {
    int p = threadIdx.x;
    if (p >= NPATH) return;
    int l1 = PATH_L1[p], l2 = PATH_L2[p], l3 = PATH_L3[p];
    int d1 = 2 * l1 + 1, d2 = 2 * l2 + 1, d3 = 2 * l3 + 1;
    double U1r[7][7], U1i[7][7], U2r[7][7], U2i[7][7], U3r[7][7], U3i[7][7];
    build_u(l1, U1r, U1i); build_u(l2, U2r, U2i); build_u(l3, U3r, U3i);
    int off = PATH_OFF[p];
    for (int a = 0; a < d1; a++)
        for (int b = 0; b < d2; b++)
            for (int c = 0; c < d3; c++) {
                double sre = 0.0;
                for (int m = 0; m < d1; m++)
                    for (int n = 0; n < d2; n++) {
                        int pp = (m - l1) + (n - l2) + l3;
                        if (pp < 0 || pp >= d3) continue;
                        double T = cg_coef(l1, m - l1, l2, n - l2, l3, pp - l3);
                        if (T == 0.0) continue;
                        double re1 = U1r[a][m], im1 = U1i[a][m];
                        double re2 = U2r[b][n], im2 = U2i[b][n];
                        double re3 = U3r[c][pp], im3 = -U3i[c][pp]; // conj
                        double rr = re1 * re2 - im1 * im2;
                        double ri = re1 * im2 + im1 * re2;
                        sre += (rr * re3 - ri * im3) * T;
                    }
                cg[off + (a * d2 + b) * d3 + c] = (float)sre;
            }
}

// ---------------- per-edge geometry: radial basis + spherical harmonics ----------------
__global__ void edge_geom_kernel(const float* __restrict__ disp,
                                 const int* __restrict__ nbr,
                                 const int* __restrict__ Z,
                                 const float* __restrict__ emb_rad,
                                 float* __restrict__ rb,
                                 float* __restrict__ sh,
                                 float* __restrict__ y0out,
                                 int nE) {
    int e = blockIdx.x * blockDim.x + threadIdx.x;
    if (e >= nE) return;
    float dx = disp[e * 3 + 0], dy = disp[e * 3 + 1], dz = disp[e * 3 + 2];
    float d = sqrtf(dx * dx + dy * dy + dz * dz);
    float inv = 1.0f / (d + 1e-9f);
    float x = dx * inv, y = dy * inv, z = dz * inv;

    const float c0 = 0.28209479177387814f, c1 = 0.4886025119029199f;
    const float c2a = 1.0925484305920792f, c2b = 0.31539156525252005f, c2c = 0.5462742152960396f;
    float s[9];
    s[0] = c0; s[1] = c1 * y; s[2] = c1 * z; s[3] = c1 * x;
    s[4] = c2a * x * y; s[5] = c2a * y * z; s[6] = c2b * (3.0f * z * z - 1.0f);
    s[7] = c2a * x * z; s[8] = c2c * (x * x - y * y);
    #pragma unroll
    for (int a = 0; a < 9; a++) sh[e * 9 + a] = s[a];

    float mask = (d < CUTOFF_R) ? 1.0f : 0.0f;
    int i_idx = nbr[e * 2 + 0];
    int j_idx = nbr[e * 2 + 1];
    int zj = Z[j_idx];
    const float PI_F = 3.14159265358979323846f;
    #pragma unroll
    for (int r = 0; r < RDIM; r++) {
        float arg = (float)(r + 1) * d * (1.0f / CUTOFF_R);
        float t = PI_F * arg;
        float sv = (t < 1e-6f) ? 1.0f : (sinf(t) / t);
        float rv = mask * sv;
        rb[e * RDIM + r] = rv;
        // y0 scatter: y[i, channel 0, f=r] += rb * emb_rad[Z[j], r]
        atomicAdd(&y0out[(size_t)i_idx * (CH * FDIM) + r], rv * emb_rad[zj * RDIM + r]);
    }
}

// ---------------- message kernel: tensor-product paths + scatter ----------------
// block = 256 threads = 16 edges x 16 features
__global__ void __launch_bounds__(256)
msg_kernel(const float* __restrict__ rb, const float* __restrict__ sh,
           const float* __restrict__ ycur,
           const float* __restrict__ Wb_l,   // 3*16*16 (this layer)
           const float* __restrict__ bb_l,   // 16
           const float* __restrict__ pw_l,   // 17*16
           const int* __restrict__ nbr,
           const float* __restrict__ cg,
           float* __restrict__ ymp, int nE) {
    __shared__ float s_cg[CG_TOTAL];
    __shared__ float s_wb[3 * RDIM * FDIM];
    __shared__ float s_pw[NPATH * FDIM];
    __shared__ float s_bb[FDIM];
    __shared__ float s_rb[16 * RDIM];
    __shared__ float s_sh[16 * 9];

    int tid = threadIdx.x;
    int e0 = blockIdx.x * 16;
    int f = tid & 15;
    int el = tid >> 4;
    int e = e0 + el;

    // Read edge endpoints early and prefetch the gather row (hide HBM latency
    // behind the LDS staging phase).  global_prefetch_b8 path (gfx1250).
    int i_idx = 0, j_idx = 0;
    if (e < nE) { i_idx = nbr[e * 2 + 0]; j_idx = nbr[e * 2 + 1]; }
    size_t jb = (size_t)j_idx * (CH * FDIM) + f;
    __builtin_prefetch(&ycur[jb], 0, 3);

    for (int t = tid; t < CG_TOTAL; t += 256) s_cg[t] = cg[t];
    for (int t = tid; t < 3 * RDIM * FDIM; t += 256) s_wb[t] = Wb_l[t];
    for (int t = tid; t < NPATH * FDIM; t += 256) s_pw[t] = pw_l[t];
    if (tid < FDIM) s_bb[tid] = bb_l[tid];
    { int ee = e0 + (tid >> 4); s_rb[tid] = (ee < nE) ? rb[(size_t)e0 * RDIM + tid] : 0.0f; }
    for (int t = tid; t < 16 * 9; t += 256) { int ee = e0 + t / 9; s_sh[t] = (ee < nE) ? sh[(size_t)e0 * 9 + t] : 0.0f; }
    __syncthreads();

    if (e >= nE) return;

    // gather x_j = y[j]  (16 channels for this feature f)
    float xj[CH];
    #pragma unroll
    for (int c = 0; c < CH; c++) xj[c] = ycur[jb + c * FDIM];

    // rbW[l][f] = sum_r rb[r] * Wb[l][r][f]
    float rw[3];
    #pragma unroll
    for (int l = 0; l < 3; l++) {
        float acc = 0.0f;
        #pragma unroll
        for (int r = 0; r < RDIM; r++) acc += s_rb[el * RDIM + r] * s_wb[l * 256 + r * 16 + f];
        rw[l] = acc;
    }
    float bvec[9];
    bvec[0] = s_sh[el * 9 + 0] * rw[0] + s_bb[f];
    #pragma unroll
    for (int a = 1; a < 4; a++) bvec[a] = s_sh[el * 9 + a] * rw[1];
    #pragma unroll
    for (int a = 4; a < 9; a++) bvec[a] = s_sh[el * 9 + a] * rw[2];

    float msg[CH];
    #pragma unroll
    for (int c = 0; c < CH; c++) msg[c] = 0.0f;

    #pragma unroll
    for (int p = 0; p < NPATH; p++) {
        const int l1 = PATH_L1[p], l2 = PATH_L2[p], l3 = PATH_L3[p];
        const int d1 = 2 * l1 + 1, d2 = 2 * l2 + 1, d3 = 2 * l3 + 1;
        const int o1 = l1 * l1, o2 = l2 * l2, o3 = l3 * l3, off = PATH_OFF[p];
        float pm[7];
        #pragma unroll
        for (int c = 0; c < 7; c++) pm[c] = 0.0f;
        #pragma unroll
        for (int a = 0; a < d1; a++) {
            #pragma unroll
            for (int b = 0; b < d2; b++) {
                float xb = xj[o1 + a] * bvec[o2 + b];
                #pragma unroll
                for (int c = 0; c < d3; c++)
                    pm[c] += s_cg[off + (a * d2 + b) * d3 + c] * xb;
            }
        }
        float pwf = s_pw[p * 16 + f];
        #pragma unroll
        for (int c = 0; c < d3; c++) msg[o3 + c] += pm[c] * pwf;
    }

    size_t ib = (size_t)i_idx * (CH * FDIM) + f;
    #pragma unroll
    for (int c = 0; c < CH; c++) atomicAdd(&ymp[ib + c * FDIM], msg[c]);
}

// ---------------- WMMA node update: y_next = per_degree_dense(ymp/norm, Wd) + bias + ymp/norm ----------------
// Templated on degree L: dl is compile-time so row->(node,m) decomposition folds to
// magic-multiply, no runtime-divisor division in the epilogue.
// Rows = flattened (node, m) for degree L. One wave does 16 rows x 16 cols.
// Branch-free operand staging: clamped unconditional b64 loads for A, LDS b64 reads for B,
// all fragments resident before a back-to-back chain of 4 WMMAs.
template <int L>
__global__ void __launch_bounds__(128)
node_update_kernel(const float* __restrict__ ymp,
                   const float* __restrict__ Wd_l,  // 4*16*16 (this layer)
                   const float* __restrict__ bd_l,  // 16
                   const float* __restrict__ normp,
                   float* __restrict__ ynext, int N) {
    constexpr int DL = 2 * L + 1;
    __shared__ float2 sBT[8 * 16];   // (k-pair, col) fragments of Wd[L]
    const float* W = Wd_l + L * 256;
    int tid = threadIdx.x;
    {   // stage B: sBT[k2*16 + c] = (W[2k2][c], W[2k2+1][c])
        int k2 = tid >> 4, c = tid & 15;
        float2 w; w.x = W[(2 * k2) * 16 + c]; w.y = W[(2 * k2 + 1) * 16 + c];
        sBT[tid] = w;
    }
    __syncthreads();

    int lane = tid & 31;
    int tile = blockIdx.x * 4 + (tid >> 5);
    int rows = N * DL;
    int ntiles = (rows + 15) >> 4;
    bool tileValid = tile < ntiles;
    if (!tileValid) tile = ntiles - 1;   // wave-uniform clamp; EXEC stays full for WMMA
    int base = tile << 4;
    int half = lane >> 4, lr = lane & 15;
    float invn = 1.0f / normp[0];

    int rA = base + lr;
    bool okA = rA < rows;
    int rC = okA ? rA : (rows - 1);            // clamped, always in-range
    int nA = rC / DL, mA = rC - nA * DL;
    size_t idxA = (size_t)nA * (CH * FDIM) + (size_t)(L * L + mA) * FDIM;
    float okf = okA ? invn : 0.0f;             // zero invalid rows arithmetically

    // issue all A loads (b64) and B reads (ds b64) up front
    const float2* ymp2 = (const float2*)ymp;
    size_t row2 = idxA >> 1;
    float2 a[4], b[4];
    #pragma unroll
    for (int s = 0; s < 4; s++) a[s] = ymp2[row2 + (2 * s + half)];
    #pragma unroll
    for (int s = 0; s < 4; s++) b[s] = sBT[(2 * s + half) * 16 + lr];

    v8f acc = {};
    #pragma unroll
    for (int s = 0; s < 4; s++) {
        v2f A = {a[s].x * okf, a[s].y * okf};
        v2f B = {b[s].x, b[s].y};
        acc = __builtin_amdgcn_wmma_f32_16x16x4_f32(false, A, false, B, (short)0, acc,
                                                    false, false);
    }

    float bias = (L == 0) ? bd_l[lr] : 0.0f;
    #pragma unroll
    for (int v = 0; v < 8; v++) {
        int r = base + v + 8 * half;
        int rcl = (r < rows) ? r : (rows - 1);
        int n = rcl / DL, m = rcl - n * DL;
        size_t idx = (size_t)n * (CH * FDIM) + (size_t)(L * L + m) * FDIM + lr;
        float res = ymp[idx] * invn;           // unconditional clamped read
        if (tileValid && r < rows) ynext[idx] = acc[v] + res + bias;
    }
}

// ---------------- WMMA species-embedding GEMM: out[:,0,:] = y[:,0,:] + emb[Z] @ Wt + bt ----------------
__global__ void __launch_bounds__(128)
embed_out_kernel(const int* __restrict__ Z,
                 const float* __restrict__ emb,   // 100 x 64
                 const float* __restrict__ Wt,    // 64 x 16
                 const float* __restrict__ bt,    // 16
                 const float* __restrict__ yfin,
                 float* __restrict__ out, int N) {
    __shared__ float2 sBT[32 * 16];  // (k-pair, col) fragments of Wt (K=64)
    int tid = threadIdx.x;
    for (int t = tid; t < 32 * 16; t += 128) {
        int k2 = t >> 4, c = t & 15;
        float2 w; w.x = Wt[(2 * k2) * 16 + c]; w.y = Wt[(2 * k2 + 1) * 16 + c];
        sBT[t] = w;
    }
    __syncthreads();

    int lane = tid & 31;
    int tile = blockIdx.x * 4 + (tid >> 5);
    int ntiles = (N + 15) >> 4;
    bool tileValid = tile < ntiles;
    if (!tileValid) tile = ntiles - 1;
    int base = tile << 4;
    int half = lane >> 4, lr = lane & 15;

    int rA = base + lr;
    bool okA = rA < N;
    int rC = okA ? rA : (N - 1);
    int zA = Z[rC];
    float okf = okA ? 1.0f : 0.0f;
    const float2* erow2 = (const float2*)(emb + (size_t)zA * 64);

    float2 a[16];
    #pragma unroll
    for (int s = 0; s < 16; s++) a[s] = erow2[2 * s + half];

    v8f acc = {};
    #pragma unroll
    for (int s = 0; s < 16; s++) {  // K = 64 in slices of 4
        float2 b = sBT[(2 * s + half) * 16 + lr];
        v2f A = {a[s].x * okf, a[s].y * okf};
        v2f B = {b.x, b.y};
        acc = __builtin_amdgcn_wmma_f32_16x16x4_f32(false, A, false, B, (short)0, acc,
                                                    false, false);
    }

    #pragma unroll
    for (int v = 0; v < 8; v++) {
        int r = base + v + 8 * half;
        if (tileValid && r < N) {
            size_t idx = (size_t)r * (CH * FDIM) + lr;  // channel 0
            out[idx] = yfin[idx] + acc[v] + bt[lr];
        }
    }
}

// ---------------- host launcher ----------------
extern "C" void kernel_launch(void* const* d_in, const int* in_sizes, int n_in,
                              void* d_out, int out_size, void* d_ws, size_t ws_size,
                              hipStream_t stream) {
    const float* disp    = (const float*)d_in[0];
    const float* emb     = (const float*)d_in[1];
    const float* emb_rad = (const float*)d_in[2];
    const float* Wt      = (const float*)d_in[3];
    const float* bt      = (const float*)d_in[4];
    const float* Wb      = (const float*)d_in[5];   // (2,3,16,16)
    const float* bb      = (const float*)d_in[6];   // (2,16)
    const float* pw      = (const float*)d_in[7];   // (2,17,16)
    const float* Wd      = (const float*)d_in[8];   // (2,4,16,16)
    const float* bd      = (const float*)d_in[9];   // (2,16)
    const float* nrm     = (const float*)d_in[10];  // (1,)
    const int*   Z       = (const int*)d_in[11];
    const int*   nbr     = (const int*)d_in[12];
    float* out = (float*)d_out;

    int N  = in_sizes[11];
    int nE = in_sizes[12] / 2;
    int nodeElems = N * CH * FDIM;

    float* ws  = (float*)d_ws;
    float* cg  = ws;
    float* rb  = ws + 1088;
    float* shb = rb + (size_t)nE * RDIM;
    float* ymp = shb + (size_t)nE * 9;
    float* yA  = ymp + (size_t)nodeElems;
    float* yB  = yA + (size_t)nodeElems;

    // init: zero y, build CG tables, per-edge geometry + y0 scatter
    fill_zero_kernel<<<(nodeElems + 255) / 256, 256, 0, stream>>>(yA, nodeElems);
    init_cg_kernel<<<1, 32, 0, stream>>>(cg);
    edge_geom_kernel<<<(nE + 255) / 256, 256, 0, stream>>>(disp, nbr, Z, emb_rad, rb, shb, yA, nE);

    float* cur = yA;
    float* nxt = yB;

    // per-degree grids: tiles over N*(2l+1) rows, 4 waves per block
    auto blocksFor = [N](int dl) {
        int tiles = (N * dl + 15) / 16;
        return (tiles + 3) / 4;
    };
    int g0 = blocksFor(1), g1 = blocksFor(3), g2 = blocksFor(5), g3 = blocksFor(7);

    for (int i = 0; i < 2; i++) {
        fill_zero_kernel<<<(nodeElems + 255) / 256, 256, 0, stream>>>(ymp, nodeElems);
        msg_kernel<<<(nE + 15) / 16, 256, 0, stream>>>(
            rb, shb, cur,
            Wb + (size_t)i * 3 * RDIM * FDIM,
            bb + (size_t)i * FDIM,
            pw + (size_t)i * NPATH * FDIM,
            nbr, cg, ymp, nE);
        const float* Wd_i = Wd + (size_t)i * 4 * FDIM * FDIM;
        const float* bd_i = bd + (size_t)i * FDIM;
        node_update_kernel<0><<<g0, 128, 0, stream>>>(ymp, Wd_i, bd_i, nrm, nxt, N);
        node_update_kernel<1><<<g1, 128, 0, stream>>>(ymp, Wd_i, bd_i, nrm, nxt, N);
        node_update_kernel<2><<<g2, 128, 0, stream>>>(ymp, Wd_i, bd_i, nrm, nxt, N);
        node_update_kernel<3><<<g3, 128, 0, stream>>>(ymp, Wd_i, bd_i, nrm, nxt, N);
        float* t = cur; cur = nxt; nxt = t;
    }

    // final: out = y; out[:,0,:] += emb[Z] @ Wt + bt
    copy_kernel<<<(nodeElems + 255) / 256, 256, 0, stream>>>(out, cur, nodeElems);
    int embTiles = (N + 15) / 16;
    embed_out_kernel<<<(embTiles + 3) / 4, 128, 0, stream>>>(Z, emb, Wt, bt, cur, out, N);
}